// Model_3650722201952
// MI455X (gfx1250) — compile-verified
//
#include <hip/hip_runtime.h>
#include <hip/hip_bf16.h>

typedef __attribute__((ext_vector_type(2))) float v2f;
typedef __attribute__((ext_vector_type(8))) float v8f;

#define N0c 500000
#define E0c 4000000
#define Rc  100000
#define S1c 100000
#define E1c 200000
#define D1c 20000
#define Hc  3

__device__ __forceinline__ float leakyf(float v, float s) { return v >= 0.f ? v : s * v; }

// Order-preserving float<->uint encoding for exact atomicMax on floats
__device__ __forceinline__ unsigned fenc(float f) {
  unsigned u = __float_as_uint(f);
  return (u & 0x80000000u) ? ~u : (u | 0x80000000u);
}
__device__ __forceinline__ float fdec(unsigned u) {
  unsigned v = (u & 0x80000000u) ? (u & 0x7FFFFFFFu) : ~u;
  return __uint_as_float(v);
}

__global__ void k_init_maxenc(unsigned* __restrict__ maxenc, int n) {
  int i = blockIdx.x * blockDim.x + threadIdx.x;
  if (i < n) maxenc[i] = 0x007FFFFFu; // fenc(-inf)
}

// --- Stage 1: hypergraph edge gather-scale-scatter (16 lanes per edge, float4) ---
__global__ void k_hyper_scatter(const float* __restrict__ x,
                                const float* __restrict__ node_norm,
                                const float* __restrict__ edge_norm,
                                const int* __restrict__ esrc,
                                const int* __restrict__ edst,
                                float* __restrict__ h) {
  long long tid = (long long)blockIdx.x * blockDim.x + threadIdx.x;
  long long e = tid >> 4;
  int part = (int)(tid & 15);
  if (e >= E0c) return;
  int s = esrc[e], d = edst[e];
  float nm = node_norm[s] * node_norm[d] * edge_norm[e];
  float4 v = ((const float4*)(x + (long long)s * 64))[part];
  float* hd = h + (long long)d * 64 + part * 4;
  atomicAdd(hd + 0, v.x * nm);
  atomicAdd(hd + 1, v.y * nm);
  atomicAdd(hd + 2, v.z * nm);
  atomicAdd(hd + 3, v.w * nm);
}

// --- Stage 2: per-graph segment sum + counts ---
__global__ void k_graph_accum(const float* __restrict__ h,
                              const int* __restrict__ node2graph,
                              float* __restrict__ gsum,
                              float* __restrict__ cnt) {
  long long tid = (long long)blockIdx.x * blockDim.x + threadIdx.x;
  long long n = tid >> 4;
  int part = (int)(tid & 15);
  if (n >= N0c) return;
  int g = node2graph[n];
  float4 v = ((const float4*)(h + n * 64))[part];
  float* gd = gsum + (long long)g * 64 + part * 4;
  atomicAdd(gd + 0, v.x);
  atomicAdd(gd + 1, v.y);
  atomicAdd(gd + 2, v.z);
  atomicAdd(gd + 3, v.w);
  if (part == 0) atomicAdd(cnt + g, 1.0f);
}

// --- Stage 3: r = leaky(gmean @ lin_w^T + lin_b), f32 WMMA 16x16x4, K=64, N=32 ---
// One wave per 16-row M tile. R = 100000 is divisible by 16 (no tail).
__global__ void k_gemm_r(const float* __restrict__ gsum,
                         const float* __restrict__ cnt,
                         const float* __restrict__ lin_w,  // [32,64] row-major
                         const float* __restrict__ lin_b,  // [32]
                         float* __restrict__ r) {          // [R,32]
  int wave = threadIdx.x >> 5;
  int lane = threadIdx.x & 31;
  int mtile = blockIdx.x * (blockDim.x >> 5) + wave;
  if (mtile >= Rc / 16) return;
  int mrow = mtile * 16 + (lane & 15);
  int koff = (lane >> 4) << 1; // lanes 16-31 hold K+2 (ISA 32-bit A 16x4 layout)
  float invc = 1.0f / fmaxf(cnt[mrow], 1.0f);
  const float* arow = gsum + (long long)mrow * 64;
  int n = lane & 15;
  v8f acc0 = {}; v8f acc1 = {};
  for (int k0 = 0; k0 < 64; k0 += 4) {
    v2f a;
    a[0] = arow[k0 + koff] * invc;
    a[1] = arow[k0 + koff + 1] * invc;
    v2f b0, b1;
    b0[0] = lin_w[n * 64 + k0 + koff];
    b0[1] = lin_w[n * 64 + k0 + koff + 1];
    b1[0] = lin_w[(n + 16) * 64 + k0 + koff];
    b1[1] = lin_w[(n + 16) * 64 + k0 + koff + 1];
    acc0 = __builtin_amdgcn_wmma_f32_16x16x4_f32(false, a, false, b0, (short)0, acc0, false, false);
    acc1 = __builtin_amdgcn_wmma_f32_16x16x4_f32(false, a, false, b1, (short)0, acc1, false, false);
  }
  int rbase = (lane >> 4) * 8; // C/D layout: lanes 16-31 hold M+8
  int col = lane & 15;
  for (int i = 0; i < 8; ++i) {
    long long row = mtile * 16 + rbase + i;
    r[row * 32 + col]      = leakyf(acc0[i] + lin_b[col], 0.01f);
    r[row * 32 + 16 + col] = leakyf(acc1[i] + lin_b[16 + col], 0.01f);
  }
}

// --- Stage 4: gathered projection  out[m,0:48] = feat[idx[m],0:32] @ W^T + bias ---
// Used for fs (feat=r, idx=src_nid) and fd (feat=node_quality, idx=dst_nid).
__global__ void k_gemm_gather48(const float* __restrict__ feat,  // [?,32]
                                const int* __restrict__ idx,    // [M]
                                const float* __restrict__ W,    // [48,32] row-major
                                const float* __restrict__ bias, // [48]
                                float* __restrict__ out,        // [M,48]
                                int M) {
  int wave = threadIdx.x >> 5;
  int lane = threadIdx.x & 31;
  int mtile = blockIdx.x * (blockDim.x >> 5) + wave;
  if (mtile >= M / 16) return;
  int mrow = mtile * 16 + (lane & 15);
  int koff = (lane >> 4) << 1;
  const float* arow = feat + (long long)idx[mrow] * 32;
  int n = lane & 15;
  v8f acc0 = {}, acc1 = {}, acc2 = {};
  for (int k0 = 0; k0 < 32; k0 += 4) {
    v2f a;
    a[0] = arow[k0 + koff];
    a[1] = arow[k0 + koff + 1];
    v2f b0, b1, b2;
    b0[0] = W[n * 32 + k0 + koff];        b0[1] = W[n * 32 + k0 + koff + 1];
    b1[0] = W[(n + 16) * 32 + k0 + koff]; b1[1] = W[(n + 16) * 32 + k0 + koff + 1];
    b2[0] = W[(n + 32) * 32 + k0 + koff]; b2[1] = W[(n + 32) * 32 + k0 + koff + 1];
    acc0 = __builtin_amdgcn_wmma_f32_16x16x4_f32(false, a, false, b0, (short)0, acc0, false, false);
    acc1 = __builtin_amdgcn_wmma_f32_16x16x4_f32(false, a, false, b1, (short)0, acc1, false, false);
    acc2 = __builtin_amdgcn_wmma_f32_16x16x4_f32(false, a, false, b2, (short)0, acc2, false, false);
  }
  int rbase = (lane >> 4) * 8;
  int col = lane & 15;
  for (int i = 0; i < 8; ++i) {
    long long row = mtile * 16 + rbase + i;
    out[row * 48 + col]      = acc0[i] + bias[col];
    out[row * 48 + 16 + col] = acc1[i] + bias[16 + col];
    out[row * 48 + 32 + col] = acc2[i] + bias[32 + col];
  }
}

// --- Stage 5: GAT edge logits + segment max (exact via encoded atomicMax) ---
__global__ void k_gat_logits(const float* __restrict__ fs, const float* __restrict__ fd,
                             const int* __restrict__ es, const int* __restrict__ ed,
                             const float* __restrict__ attn,
                             float* __restrict__ logits, unsigned* __restrict__ maxenc) {
  int tid = blockIdx.x * blockDim.x + threadIdx.x;
  if (tid >= E1c * Hc) return;
  int e = tid / Hc, hh = tid % Hc;
  int s = es[e], d = ed[e];
  const float* fsr = fs + (long long)s * 48 + hh * 16;
  const float* fdr = fd + (long long)d * 48 + hh * 16;
  const float* at = attn + hh * 16;
  float acc = 0.f;
#pragma unroll
  for (int o = 0; o < 16; ++o) acc += at[o] * leakyf(fsr[o] + fdr[o], 0.2f);
  logits[tid] = acc;
  atomicMax(maxenc + d * Hc + hh, fenc(acc));
}

// --- Stage 6: exp(logit - max) + segment denominator ---
__global__ void k_gat_exp(const float* __restrict__ logits, const unsigned* __restrict__ maxenc,
                          const int* __restrict__ ed,
                          float* __restrict__ ex, float* __restrict__ den) {
  int tid = blockIdx.x * blockDim.x + threadIdx.x;
  if (tid >= E1c * Hc) return;
  int e = tid / Hc, hh = tid % Hc;
  int d = ed[e];
  float mx = fdec(maxenc[d * Hc + hh]);
  if (!isfinite(mx)) mx = 0.f;
  float v = __expf(logits[tid] - mx);
  ex[tid] = v;
  atomicAdd(den + d * Hc + hh, v);
}

// --- Stage 7: out[d,o] += sum_h alpha[e,h] * fs[src,h,o]  (heads fused, 1 atomic per (e,o)) ---
__global__ void k_gat_out(const float* __restrict__ fs, const float* __restrict__ ex,
                          const float* __restrict__ den,
                          const int* __restrict__ es, const int* __restrict__ ed,
                          float* __restrict__ out) {
  long long tid = (long long)blockIdx.x * blockDim.x + threadIdx.x;
  if (tid >= (long long)E1c * 16) return;
  int e = (int)(tid >> 4);
  int o = (int)(tid & 15);
  int s = es[e], d = ed[e];
  float acc = 0.f;
#pragma unroll
  for (int hh = 0; hh < Hc; ++hh) {
    float alpha = ex[e * Hc + hh] / fmaxf(den[d * Hc + hh], 1e-9f);
    acc += alpha * fs[(long long)s * 48 + hh * 16 + o];
  }
  atomicAdd(out + (long long)d * 16 + o, acc);
}

extern "C" void kernel_launch(void* const* d_in, const int* in_sizes, int n_in,
                              void* d_out, int out_size, void* d_ws, size_t ws_size,
                              hipStream_t stream) {
  const float* x          = (const float*)d_in[0];
  const float* node_norm  = (const float*)d_in[1];
  const float* edge_norm  = (const float*)d_in[2];
  const float* lin_w      = (const float*)d_in[3];
  const float* lin_b      = (const float*)d_in[4];
  const float* w_src      = (const float*)d_in[5];
  const float* b_src      = (const float*)d_in[6];
  const float* w_dst      = (const float*)d_in[7];
  const float* b_dst      = (const float*)d_in[8];
  const float* attn       = (const float*)d_in[9];
  const float* node_qual  = (const float*)d_in[10];
  const int*   edge_src   = (const int*)d_in[11];
  const int*   edge_dst   = (const int*)d_in[12];
  const int*   node2graph = (const int*)d_in[13];
  const int*   src_nid    = (const int*)d_in[14];
  const int*   dst_nid    = (const int*)d_in[15];
  const int*   e1_src     = (const int*)d_in[16];
  const int*   e1_dst     = (const int*)d_in[17];

  // Workspace layout; zero-init region is placed first and cleared with one memset.
  char* ws = (char*)d_ws;
  size_t off = 0;
  auto take = [&](size_t bytes) -> void* {
    void* p = ws + off;
    off = (off + bytes + 255) & ~(size_t)255;
    return p;
  };
  float*    h      = (float*)take((size_t)N0c * 64 * 4);
  float*    gsum   = (float*)take((size_t)Rc * 64 * 4);
  float*    cnt    = (float*)take((size_t)Rc * 4);
  float*    den    = (float*)take((size_t)D1c * Hc * 4);
  size_t zero_bytes = off; // everything above must start at 0
  float*    r      = (float*)take((size_t)Rc * 32 * 4);
  float*    fs     = (float*)take((size_t)S1c * 48 * 4);
  float*    fd     = (float*)take((size_t)D1c * 48 * 4);
  float*    logits = (float*)take((size_t)E1c * Hc * 4);
  float*    ex     = (float*)take((size_t)E1c * Hc * 4);
  unsigned* maxenc = (unsigned*)take((size_t)D1c * Hc * 4);
  (void)ws_size; (void)in_sizes; (void)n_in;

  hipMemsetAsync(d_ws, 0, zero_bytes, stream);
  hipMemsetAsync(d_out, 0, (size_t)out_size * sizeof(float), stream);

  const int B = 256;
  k_init_maxenc<<<(D1c * Hc + B - 1) / B, B, 0, stream>>>(maxenc, D1c * Hc);

  long long t0 = (long long)E0c * 16;
  k_hyper_scatter<<<(unsigned)((t0 + B - 1) / B), B, 0, stream>>>(
      x, node_norm, edge_norm, edge_src, edge_dst, h);

  long long t1 = (long long)N0c * 16;
  k_graph_accum<<<(unsigned)((t1 + B - 1) / B), B, 0, stream>>>(h, node2graph, gsum, cnt);

  // WMMA GEMMs: 8 waves/block, 1 wave per 16-row M tile
  int mt_r = Rc / 16;
  k_gemm_r<<<(mt_r + 7) / 8, B, 0, stream>>>(gsum, cnt, lin_w, lin_b, r);

  int mt_fs = S1c / 16;
  k_gemm_gather48<<<(mt_fs + 7) / 8, B, 0, stream>>>(r, src_nid, w_src, b_src, fs, S1c);
  int mt_fd = D1c / 16;
  k_gemm_gather48<<<(mt_fd + 7) / 8, B, 0, stream>>>(node_qual, dst_nid, w_dst, b_dst, fd, D1c);

  int t2 = E1c * Hc;
  k_gat_logits<<<(t2 + B - 1) / B, B, 0, stream>>>(fs, fd, e1_src, e1_dst, attn, logits, maxenc);
  k_gat_exp<<<(t2 + B - 1) / B, B, 0, stream>>>(logits, maxenc, e1_dst, ex, den);

  long long t3 = (long long)E1c * 16;
  k_gat_out<<<(unsigned)((t3 + B - 1) / B), B, 0, stream>>>(
      fs, ex, den, e1_src, e1_dst, (float*)d_out);
}